// Model_80985903333894
// MI455X (gfx1250) — compile-verified
//
#include <hip/hip_runtime.h>
#include <hip/hip_bf16.h>

// ---------------------------------------------------------------------------
// Problem constants (from reference)
// ---------------------------------------------------------------------------
#define Mn     8192      // nodes
#define W_IN   64
#define HID    32
#define BSZ    16
#define NNZ    131072
#define EXTRA  32
#define KMAIN  (Mn * HID)          // 262144
#define KTOT   (KMAIN + EXTRA)     // 262176
#define NFC1   512

typedef float v2f __attribute__((ext_vector_type(2)));
typedef float v8f __attribute__((ext_vector_type(8)));

// ---------------------------------------------------------------------------
// Utility kernels
// ---------------------------------------------------------------------------
__global__ void k_zero(float* __restrict__ p, int n) {
    int i = blockIdx.x * blockDim.x + threadIdx.x;
    if (i < n) p[i] = 0.0f;
}

__global__ void k_hist(const int* __restrict__ rows, int* __restrict__ counts, int nnz) {
    int i = blockIdx.x * blockDim.x + threadIdx.x;
    if (i < nnz) atomicAdd(&counts[rows[i]], 1);
}

// Exclusive prefix scan of 8192 ints in one 1024-thread block (8 per thread).
__global__ void k_scan(int* __restrict__ counts) {
    __shared__ int lds[1024];
    int t = threadIdx.x;
    int local[8];
    int s = 0;
#pragma unroll
    for (int i = 0; i < 8; ++i) { local[i] = counts[t * 8 + i]; s += local[i]; }
    lds[t] = s;
    __syncthreads();
    for (int off = 1; off < 1024; off <<= 1) {
        int v = (t >= off) ? lds[t - off] : 0;
        __syncthreads();
        lds[t] += v;
        __syncthreads();
    }
    int run = lds[t] - s;   // exclusive offset for this thread's chunk
#pragma unroll
    for (int i = 0; i < 8; ++i) { int d = local[i]; counts[t * 8 + i] = run; run += d; }
    if (t == 1023) counts[Mn] = run;
}

__global__ void k_scatter(const int* __restrict__ rows, const int* __restrict__ cols,
                          const float* __restrict__ vals, const int* __restrict__ row_ptr,
                          int* __restrict__ fill, int* __restrict__ ccol,
                          float* __restrict__ cval, int nnz) {
    int i = blockIdx.x * blockDim.x + threadIdx.x;
    if (i >= nnz) return;
    int r = rows[i];
    int pos = row_ptr[r] + atomicAdd(&fill[r], 1);
    ccol[pos] = cols[i];
    cval[pos] = vals[i];
}

// ---------------------------------------------------------------------------
// CSR SpMM: lx[m, j] = sum_e val[e] * xin[b, col[e], f]   (j = b*FW + f)
// One wave per (row m, 128-feature chunk); float4 gathers.
// ---------------------------------------------------------------------------
__global__ void k_spmm(const int* __restrict__ row_ptr, const int* __restrict__ ccol,
                       const float* __restrict__ cval, const float* __restrict__ xin,
                       float* __restrict__ lx, int fw_shift) {
    int m = blockIdx.x;
    int j = blockIdx.y * 128 + threadIdx.x * 4;       // feature index, 4-wide
    int b = j >> fw_shift;
    int f = j & ((1 << fw_shift) - 1);
    int e0 = row_ptr[m], e1 = row_ptr[m + 1];
    float4 acc = make_float4(0.f, 0.f, 0.f, 0.f);
    const float* xb = xin + (((size_t)b << 13) << fw_shift) + f;   // b*Mn*FW + f
    for (int e = e0; e < e1; ++e) {
        int   c = ccol[e];
        float v = cval[e];
        float4 xv = *(const float4*)(xb + ((size_t)c << fw_shift));
        acc.x += v * xv.x; acc.y += v * xv.y; acc.z += v * xv.z; acc.w += v * xv.w;
    }
    *(float4*)(lx + (size_t)m * (BSZ << fw_shift) + j) = acc;
}

// ---------------------------------------------------------------------------
// WMMA f32 16x16x4 Linear: out(b,m,c) = A(r=m*16+b, 0..K-1) @ W(K,32) + bias
// One wave per (node m = row tile, 16-col tile).
// ---------------------------------------------------------------------------
__global__ void k_lin_wmma(const float* __restrict__ A, const float* __restrict__ W,
                           const float* __restrict__ bias, float* __restrict__ out, int K) {
    int rt = blockIdx.x;           // node m
    int ct = blockIdx.y;           // 0..1
    int l  = threadIdx.x;
    int half = l >> 4, lr = l & 15;
    const float* arow = A + (size_t)(rt * 16 + lr) * K;
    int col = ct * 16 + lr;
    v8f c = {};
    for (int k0 = 0; k0 < K; k0 += 4) {
        int ka = k0 + half * 2;
        v2f a, b;
        a.x = arow[ka];            a.y = arow[ka + 1];
        b.x = W[ka * HID + col];   b.y = W[(ka + 1) * HID + col];
        c = __builtin_amdgcn_wmma_f32_16x16x4_f32(false, a, false, b, (short)0, c, false, false);
    }
    float bi = bias[col];
#pragma unroll
    for (int v = 0; v < 8; ++v) {
        int bidx = v + half * 8;   // batch index (row-in-tile)
        out[((size_t)bidx * Mn + rt) * HID + col] = c[v] + bi;
    }
}

// ---------------------------------------------------------------------------
// Per-channel (32) sum / sumsq reduction: two-stage (LDS then 64 global atomics).
// ---------------------------------------------------------------------------
__global__ void k_stats(const float* __restrict__ x, float* __restrict__ sums, int n) {
    __shared__ float ssum[HID], ssq[HID];
    int t = threadIdx.x;
    if (t < HID) { ssum[t] = 0.f; ssq[t] = 0.f; }
    __syncthreads();
    int ch = t & (HID - 1);        // stride is a multiple of 32 -> channel fixed per thread
    float ls = 0.f, lq = 0.f;
    for (int i = blockIdx.x * blockDim.x + t; i < n; i += gridDim.x * blockDim.x) {
        float v = x[i];
        ls += v; lq += v * v;
    }
    atomicAdd(&ssum[ch], ls);
    atomicAdd(&ssq[ch], lq);
    __syncthreads();
    if (t < HID) {
        atomicAdd(&sums[t], ssum[t]);
        atomicAdd(&sums[HID + t], ssq[t]);
    }
}

__global__ void k_bnrelu(float* __restrict__ x, const float* __restrict__ sums,
                         const float* __restrict__ g, const float* __restrict__ be,
                         int n, float invN) {
    int i = blockIdx.x * blockDim.x + threadIdx.x;
    if (i >= n) return;
    int ch = i & (HID - 1);
    float mean = sums[ch] * invN;
    float var  = sums[HID + ch] * invN - mean * mean;  // biased variance (training BN)
    float v = g[ch] * (x[i] - mean) * rsqrtf(var + 1e-5f) + be[ch];
    x[i] = fmaxf(v, 0.f);
}

// ---------------------------------------------------------------------------
// fc1: h(16,512) += X(16, 262176) @ W(262176, 512), WMMA f32 split-K.
// Each wave owns a 32-column window (two 16x16 accumulators sharing one A
// fragment) so every streamed W row is a full 128B cacheline per wave.
// gridDim = (16 n-groups, 256 k-slabs of 1024); last slab also covers the
// 32-element ex_inputs tail via a branch-free tail loop.
// ---------------------------------------------------------------------------
__global__ void k_fc1_wmma(const float* __restrict__ X,   // x2 flat: (16, 262144)
                           const float* __restrict__ EX,  // (16, 32)
                           const float* __restrict__ W,   // (262176, 512)
                           float* __restrict__ h) {
    int ng = blockIdx.x;                  // 0..15: 32-column group
    int ks = blockIdx.y;                  // 0..255: K-slab
    int l  = threadIdx.x;
    int half = l >> 4, lr = l & 15;
    int n0 = ng * 32 + lr;                // columns n0 (tile0) and n0+16 (tile1)
    int kbeg = ks * 1024;
    int kend = (ks == (int)gridDim.y - 1) ? KTOT : kbeg + 1024;
    int kmid = kend < KMAIN ? kend : KMAIN;
    const float* xrow = X + (size_t)lr * KMAIN;           // A row = batch = lr
    v8f c0 = {}, c1 = {};
    int k0 = kbeg;
#pragma unroll 4
    for (; k0 < kmid; k0 += 4) {
        int ka = k0 + half * 2;
        const float* wr0 = W + (size_t)ka * NFC1 + n0;
        const float* wr1 = wr0 + NFC1;
        v2f a, b0, b1;
        a.x  = xrow[ka];  a.y  = xrow[ka + 1];
        b0.x = wr0[0];    b0.y = wr1[0];
        b1.x = wr0[16];   b1.y = wr1[16];
        c0 = __builtin_amdgcn_wmma_f32_16x16x4_f32(false, a, false, b0, (short)0, c0, false, false);
        c1 = __builtin_amdgcn_wmma_f32_16x16x4_f32(false, a, false, b1, (short)0, c1, false, false);
    }
    for (; k0 < kend; k0 += 4) {          // ex_inputs tail (k >= KMAIN only)
        int ka = k0 + half * 2;
        const float* wr0 = W + (size_t)ka * NFC1 + n0;
        const float* wr1 = wr0 + NFC1;
        v2f a, b0, b1;
        a.x  = EX[lr * EXTRA + (ka - KMAIN)];
        a.y  = EX[lr * EXTRA + (ka + 1 - KMAIN)];
        b0.x = wr0[0];    b0.y = wr1[0];
        b1.x = wr0[16];   b1.y = wr1[16];
        c0 = __builtin_amdgcn_wmma_f32_16x16x4_f32(false, a, false, b0, (short)0, c0, false, false);
        c1 = __builtin_amdgcn_wmma_f32_16x16x4_f32(false, a, false, b1, (short)0, c1, false, false);
    }
#pragma unroll
    for (int v = 0; v < 8; ++v) {
        int bidx = v + half * 8;          // batch row
        atomicAdd(&h[bidx * NFC1 + n0],      c0[v]);
        atomicAdd(&h[bidx * NFC1 + n0 + 16], c1[v]);
    }
}

// ---------------------------------------------------------------------------
// head: out(16,2) = relu(h + fc1_b) @ fc2_w + fc2_b
// ---------------------------------------------------------------------------
__global__ void k_head(const float* __restrict__ h, const float* __restrict__ fc1b,
                       const float* __restrict__ w2, const float* __restrict__ b2,
                       float* __restrict__ out) {
    int t = threadIdx.x;               // 32 threads: (b, c) pairs
    int b = t >> 1, c = t & 1;
    float acc = b2[c];
    for (int j = 0; j < NFC1; ++j) {
        float hv = fmaxf(h[b * NFC1 + j] + fc1b[j], 0.f);
        acc += hv * w2[j * 2 + c];
    }
    out[b * 2 + c] = acc;
}

// ---------------------------------------------------------------------------
// Launch
// ---------------------------------------------------------------------------
extern "C" void kernel_launch(void* const* d_in, const int* in_sizes, int n_in,
                              void* d_out, int out_size, void* d_ws, size_t ws_size,
                              hipStream_t stream) {
    const float* inputs    = (const float*)d_in[0];   // (16, 8192, 64)
    const float* ex_inputs = (const float*)d_in[1];   // (16, 32)
    const int*   edge_row  = (const int*)  d_in[2];
    const int*   edge_col  = (const int*)  d_in[3];
    const float* edge_val  = (const float*)d_in[4];
    const float* W1  = (const float*)d_in[5];
    const float* b1  = (const float*)d_in[6];
    const float* g1  = (const float*)d_in[7];
    const float* be1 = (const float*)d_in[8];
    const float* W2  = (const float*)d_in[9];
    const float* b2  = (const float*)d_in[10];
    const float* g2  = (const float*)d_in[11];
    const float* be2 = (const float*)d_in[12];
    const float* fc1w = (const float*)d_in[13];
    const float* fc1b = (const float*)d_in[14];
    const float* fc2w = (const float*)d_in[15];
    const float* fc2b = (const float*)d_in[16];
    float* out = (float*)d_out;

    // Workspace layout (256B-aligned offsets)
    char* ws = (char*)d_ws;
    int*   counts = (int*)  (ws + 0);          // (Mn+1) ints -> row_ptr after scan
    int*   fill   = (int*)  (ws + 33024);      // Mn ints
    int*   ccol   = (int*)  (ws + 65792);      // NNZ ints
    float* cval   = (float*)(ws + 590080);     // NNZ floats
    float* stats  = (float*)(ws + 1114368);    // 128 floats (layer1: [0,64), layer2: [64,128))
    float* h      = (float*)(ws + 1114880);    // 16*512 floats
    float* Lx     = (float*)(ws + 1147648);    // up to 8192*1024 floats (32 MB)
    float* x1     = (float*)(ws + 34702080);   // 16*8192*32 floats (16 MB)
    float* x2     = (float*)(ws + 51479296);   // 16*8192*32 floats (16 MB)

    const int NELEM = BSZ * Mn * HID;          // 4194304
    const float invN = 1.0f / (float)(BSZ * Mn);

    // --- init ---
    k_zero<<<(Mn + 1 + 255) / 256, 256, 0, stream>>>((float*)counts, Mn + 1);
    k_zero<<<(Mn + 255) / 256, 256, 0, stream>>>((float*)fill, Mn);
    k_zero<<<1, 128, 0, stream>>>(stats, 128);
    k_zero<<<32, 256, 0, stream>>>(h, BSZ * NFC1);

    // --- CSR build (shared by both layers) ---
    k_hist<<<NNZ / 256, 256, 0, stream>>>(edge_row, counts, NNZ);
    k_scan<<<1, 1024, 0, stream>>>(counts);
    k_scatter<<<NNZ / 256, 256, 0, stream>>>(edge_row, edge_col, edge_val,
                                             counts, fill, ccol, cval, NNZ);

    // --- layer 1 ---
    k_spmm<<<dim3(Mn, (BSZ * W_IN) / 128), 32, 0, stream>>>(counts, ccol, cval, inputs, Lx, 6);
    k_lin_wmma<<<dim3(Mn, 2), 32, 0, stream>>>(Lx, W1, b1, x1, W_IN);
    k_stats<<<4096, 256, 0, stream>>>(x1, stats, NELEM);
    k_bnrelu<<<NELEM / 256, 256, 0, stream>>>(x1, stats, g1, be1, NELEM, invN);

    // --- layer 2 ---
    k_spmm<<<dim3(Mn, (BSZ * HID) / 128), 32, 0, stream>>>(counts, ccol, cval, x1, Lx, 5);
    k_lin_wmma<<<dim3(Mn, 2), 32, 0, stream>>>(Lx, W2, b2, x2, HID);
    k_stats<<<4096, 256, 0, stream>>>(x2, stats + 64, NELEM);
    k_bnrelu<<<NELEM / 256, 256, 0, stream>>>(x2, stats + 64, g2, be2, NELEM, invN);

    // --- head ---
    k_fc1_wmma<<<dim3(NFC1 / 32, 256), 32, 0, stream>>>(x2, ex_inputs, fc1w, h);
    k_head<<<1, 32, 0, stream>>>(h, fc1b, fc2w, fc2b, out);
}